// ModelNew_72902774882654
// MI455X (gfx1250) — compile-verified
//
#include <hip/hip_runtime.h>
#include <math.h>

// Fused top-k -> top-p -> gumbel sampling for [256, 128000] f32 logits.
// The kept set in sorted space is always a prefix => per-row scalar threshold.
// Kernel 1: radix histogram + top-64 selection, with a DEPTH-8 CDNA5 async
//           global->LDS pipeline (global_load_async_to_lds_b128 / ASYNCcnt)
//           so each tile iteration costs ~latency/8 instead of full latency.
// Kernel 2: streams logits (L2-resident, 131MB < 192MB L2) + q (NT load),
//           writes masked logits (NT store), packed atomic argmax of
//           l - log(q+eps) among kept elements.
// Kernel 3: unpacks winner indices.

#define ROWS     256
#define VOCAB    128000
#define TILE_F   1024                 // floats per staged tile (256 thr * 4)
#define NTILES   (VOCAB / TILE_F)     // 125, exact
#define NBUF     8                    // async pipeline depth (power of 2)
#define BINS     2048                 // top 11 bits of ordered key
#define CAND_MAX 2048
#define TOPK     64
#define EPSF     1e-8f

typedef float v4f __attribute__((ext_vector_type(4)));

__device__ __forceinline__ unsigned f2ord(float f) {
  // monotone float -> uint mapping (larger float => larger uint)
  unsigned u = __float_as_uint(f);
  return (u & 0x80000000u) ? ~u : (u | 0x80000000u);
}

__device__ __forceinline__ unsigned lds_byte_addr(const void* p) {
  // low 32 bits of the generic (flat) address of an LDS object = LDS offset
  return (unsigned)(unsigned long long)p;
}

// CDNA5 async global -> LDS copy, 16B per lane, tracked by ASYNCcnt.
__device__ __forceinline__ void async_copy_b128(unsigned lds, const float* g) {
  asm volatile("global_load_async_to_lds_b128 %0, %1, off"
               :: "v"(lds), "v"(g)
               : "memory");
}

template <int N>
__device__ __forceinline__ void wait_asynccnt() {
  asm volatile("s_wait_asynccnt %0" :: "n"(N) : "memory");
}

// ---------------------------------------------------------------------------
// Kernel 1: per-row top-64 selection + softmax top-p cutoff -> threshold
// ---------------------------------------------------------------------------
__global__ __launch_bounds__(256) void topkp_threshold_kernel(
    const float* __restrict__ logits, const int* __restrict__ top_ks,
    const float* __restrict__ top_ps, float* __restrict__ thresholds,
    unsigned long long* __restrict__ rowBest) {
  __shared__ float    stage[NBUF][TILE_F];      // 32 KB
  __shared__ unsigned hist[BINS];               //  8 KB
  __shared__ float    cand[CAND_MAX];           //  8 KB
  __shared__ unsigned chunkSum[256];
  __shared__ float    sortedTop[TOPK];
  __shared__ unsigned sBin, sCnt;

  const int    row  = blockIdx.x;
  const int    tid  = threadIdx.x;
  const float* rowp = logits + (size_t)row * VOCAB;

  for (int i = tid; i < BINS; i += 256) hist[i] = 0u;
  if (tid == 0) sCnt = 0u;
  __syncthreads();

  const unsigned ldsBase = lds_byte_addr(&stage[0][tid * 4]);

  // ---- Pass A: histogram of top-11 key bits, depth-NBUF async pipeline ----
#pragma unroll
  for (int t = 0; t < NBUF; ++t)
    async_copy_b128(ldsBase + t * TILE_F * 4,
                    rowp + (size_t)t * TILE_F + tid * 4);
  int t = 0;
  for (; t < NTILES - NBUF; ++t) {
    wait_asynccnt<NBUF - 1>();                   // oldest tile resident
    const int buf = t & (NBUF - 1);
#pragma unroll
    for (int e = 0; e < 4; ++e) {
      unsigned key = f2ord(stage[buf][tid * 4 + e]);
      atomicAdd(&hist[key >> 21], 1u);           // ds_add
    }
    async_copy_b128(ldsBase + buf * TILE_F * 4,
                    rowp + (size_t)(t + NBUF) * TILE_F + tid * 4);
  }
  wait_asynccnt<0>();                            // drain tail (imm-only wait)
  for (; t < NTILES; ++t) {
    const int buf = t & (NBUF - 1);
#pragma unroll
    for (int e = 0; e < 4; ++e) {
      unsigned key = f2ord(stage[buf][tid * 4 + e]);
      atomicAdd(&hist[key >> 21], 1u);
    }
  }
  __syncthreads();

  // ---- Find bin b: count(key>>21 > b) < 64 <= count(key>>21 >= b) ----
  unsigned cs = 0;
#pragma unroll
  for (int i = 0; i < 8; ++i) cs += hist[tid * 8 + i];
  chunkSum[tid] = cs;
  __syncthreads();
  if (tid == 0) {
    unsigned acc = 0;
    int c = 255;
    for (; c >= 0; --c) {
      unsigned n = chunkSum[c];
      if (acc + n >= TOPK) break;
      acc += n;
    }
    int b = (c < 0) ? 0 : c * 8 + 7;
    if (c >= 0) {
      for (; b > c * 8; --b) {
        unsigned n = hist[b];
        if (acc + n >= TOPK) break;
        acc += n;
      }
    }
    sBin = (unsigned)b;
  }
  __syncthreads();
  const unsigned binThr = sBin;

  // ---- Pass B: collect candidates (key>>21 >= binThr), same pipeline ----
#pragma unroll
  for (int u = 0; u < NBUF; ++u)
    async_copy_b128(ldsBase + u * TILE_F * 4,
                    rowp + (size_t)u * TILE_F + tid * 4);
  t = 0;
  for (; t < NTILES - NBUF; ++t) {
    wait_asynccnt<NBUF - 1>();
    const int buf = t & (NBUF - 1);
#pragma unroll
    for (int e = 0; e < 4; ++e) {
      float v = stage[buf][tid * 4 + e];
      if ((f2ord(v) >> 21) >= binThr) {
        unsigned pos = atomicAdd(&sCnt, 1u);
        if (pos < CAND_MAX) cand[pos] = v;
      }
    }
    async_copy_b128(ldsBase + buf * TILE_F * 4,
                    rowp + (size_t)(t + NBUF) * TILE_F + tid * 4);
  }
  wait_asynccnt<0>();
  for (; t < NTILES; ++t) {
    const int buf = t & (NBUF - 1);
#pragma unroll
    for (int e = 0; e < 4; ++e) {
      float v = stage[buf][tid * 4 + e];
      if ((f2ord(v) >> 21) >= binThr) {
        unsigned pos = atomicAdd(&sCnt, 1u);
        if (pos < CAND_MAX) cand[pos] = v;
      }
    }
  }
  __syncthreads();

  // ---- Wave 0: selection-sort the top-64 candidates (wave32 shuffles) ----
  const int candN = (int)(sCnt < (unsigned)CAND_MAX ? sCnt : (unsigned)CAND_MAX);
  if (tid < 32) {
    for (int j = 0; j < TOPK; ++j) {
      float best = -INFINITY;
      int   bi   = -1;
      for (int i = tid; i < candN; i += 32) {
        float v = cand[i];
        if (v > best) { best = v; bi = i; }
      }
#pragma unroll
      for (int off = 16; off > 0; off >>= 1) {
        float ov = __shfl_xor(best, off);
        int   oi = __shfl_xor(bi, off);
        if (ov > best || (ov == best && oi >= 0 && (bi < 0 || oi < bi))) {
          best = ov;
          bi   = oi;
        }
      }
      if (tid == 0) {
        sortedTop[j] = best;
        if (bi >= 0) cand[bi] = -INFINITY;
      }
    }
  }
  __syncthreads();

  // ---- Thread 0: softmax over top-k, exclusive-cumsum top-p cutoff ----
  if (tid == 0) {
    int k = top_ks[row];
    if (k <= 0 || k > TOPK) k = TOPK;   // k<=0 disables top-k; setup has k<=63
    const float p = top_ps[row];
    const float m = sortedTop[0];
    float s = 0.f;
    for (int i = 0; i < k; ++i) s += expf(sortedTop[i] - m);
    const float lim = p * s;            // compare unnormalized excl. cumsum
    float cum = 0.f;
    int   n   = 1;                      // rank 0 always kept
    for (int i = 0; i < k; ++i) {
      if (sortedTop[i] == -INFINITY) break;
      if (i > 0) {
        if (cum <= lim) n = i + 1; else break;
      }
      cum += expf(sortedTop[i] - m);
    }
    thresholds[row] = sortedTop[n - 1];
    rowBest[row]    = 0ull;             // re-init every call (graph replay safe)
  }
}

// ---------------------------------------------------------------------------
// Kernel 2: mask logits by threshold, write selected_logits (NT), packed
// argmax of (l - log(q+eps)) among kept elements. q is read NT so the
// L2-resident logits lines are not displaced.
// ---------------------------------------------------------------------------
__global__ __launch_bounds__(256) void mask_sample_kernel(
    const float* __restrict__ logits, const float* __restrict__ q,
    const float* __restrict__ thresholds, float* __restrict__ outLogits,
    unsigned long long* __restrict__ rowBest) {
  const int    row  = blockIdx.y;
  const int    tid  = threadIdx.x;
  const int    col0 = blockIdx.x * 1024 + tid * 4;            // 125*1024 = VOCAB
  const size_t base = (size_t)row * VOCAB + col0;
  const float  thr  = thresholds[row];

  v4f l4 = *(const v4f*)(logits + base);                      // RT: L2-resident
  v4f q4 = __builtin_nontemporal_load((const v4f*)(q + base));// NT: read once

  float bestScore = -INFINITY;
  int   bestIdx   = -1;
  v4f   o4;
#pragma unroll
  for (int e = 0; e < 4; ++e) {
    const float lv   = l4[e];
    const bool  keep = lv >= thr;
    o4[e] = keep ? lv : -INFINITY;
    if (keep) {
      float sc = lv - logf(q4[e] + EPSF);
      if (sc > bestScore) { bestScore = sc; bestIdx = col0 + e; }
    }
  }
  __builtin_nontemporal_store(o4, (v4f*)(outLogits + base));  // NT: write once

  unsigned long long key =
      (bestIdx < 0)
          ? 0ull
          : ((unsigned long long)f2ord(bestScore) << 32) |
                (unsigned)(0x7FFFFFFF - bestIdx);   // smaller idx wins ties

  __shared__ unsigned long long red[256];
  red[tid] = key;
  __syncthreads();
  for (int s2 = 128; s2 > 0; s2 >>= 1) {
    if (tid < s2) {
      unsigned long long o2 = red[tid + s2];
      if (o2 > red[tid]) red[tid] = o2;
    }
    __syncthreads();
  }
  if (tid == 0 && red[0] != 0ull) atomicMax(rowBest + row, red[0]);
}

// ---------------------------------------------------------------------------
// Kernel 3: unpack per-row winner index into d_out[0:256]
// ---------------------------------------------------------------------------
__global__ void finalize_idx_kernel(const unsigned long long* __restrict__ rowBest,
                                    float* __restrict__ outIdx) {
  const int row = blockIdx.x * blockDim.x + threadIdx.x;
  if (row < ROWS) {
    unsigned long long k = rowBest[row];
    int idx = (k == 0ull) ? 0 : (0x7FFFFFFF - (int)(unsigned)(k & 0xFFFFFFFFull));
    outIdx[row] = (float)idx;   // int32 index, exact in f32 (< 2^24)
  }
}

extern "C" void kernel_launch(void* const* d_in, const int* in_sizes, int n_in,
                              void* d_out, int out_size, void* d_ws, size_t ws_size,
                              hipStream_t stream) {
  const float* logits = (const float*)d_in[0];
  const int*   top_ks = (const int*)d_in[1];
  const float* top_ps = (const float*)d_in[2];
  const float* q      = (const float*)d_in[3];

  float* outIdx    = (float*)d_out;           // selected_idx (256)
  float* outLogits = (float*)d_out + ROWS;    // selected_logits (256*128000)

  float*              thresholds = (float*)d_ws;                       // 1 KB
  unsigned long long* rowBest    = (unsigned long long*)((char*)d_ws + 2048);

  topkp_threshold_kernel<<<ROWS, 256, 0, stream>>>(logits, top_ks, top_ps,
                                                   thresholds, rowBest);
  dim3 g2(VOCAB / 1024, ROWS);                // (125, 256)
  mask_sample_kernel<<<g2, 256, 0, stream>>>(logits, q, thresholds, outLogits,
                                             rowBest);
  finalize_idx_kernel<<<1, 256, 0, stream>>>(rowBest, outIdx);
}